// DeformableTransformer_18975165514252
// MI455X (gfx1250) — compile-verified
//
#include <hip/hip_runtime.h>
#include <hip/hip_bf16.h>

// ---------------------------------------------------------------------------
// MS-Deformable-Attention for MI455X (gfx1250, wave32, WMMA).
// fp32 end-to-end; GEMMs use V_WMMA_F32_16X16X4_F32 (CDNA5 fp32 matrix op).
// Each wave computes 4 N-tiles (16x64) -> 4 independent WMMA accumulator
// chains (ILP for the matrix pipe) and 4x reuse of each A fragment.
// ---------------------------------------------------------------------------

typedef float v2f __attribute__((ext_vector_type(2)));
typedef float v8f __attribute__((ext_vector_type(8)));

#define NQ_TOTAL 43520      // N * Lq = 2 * 21760
#define GEMM_K   256        // C
#define LQ       21760
#define LDSK     258        // padded K-stride in LDS (even -> b64-aligned)
#define NTILES   4          // 16-wide N-tiles per wave

// ---------------------------------------------------------------------------
// C[M x Ncols] = A[M x 256] @ W[256 x Ncols] + bias.
// Block: 256 threads (8 waves). Block tile: 128 rows x 64 cols.
// Wave tile: 16 rows x 64 cols (4 WMMA tiles, shared A fragment).
// B strip (256 x 64 fp32, 66KB) staged in LDS transposed: lds[n][k].
// ---------------------------------------------------------------------------
__launch_bounds__(256)
__global__ void gemm_wmma_f32(const float* __restrict__ A,
                              const float* __restrict__ Wm,
                              const float* __restrict__ bias,
                              float* __restrict__ C,
                              int Ncols)
{
    __shared__ float ldsB[64 * LDSK];

    const int t  = threadIdx.x;
    const int nb = blockIdx.y * 64;          // column base of this block

    // Cooperative load of W[k][nb..nb+63] into lds[n][k] (thread t owns k-row t).
    const float* wrow = Wm + (size_t)t * Ncols + nb;
#pragma unroll
    for (int n = 0; n < 64; ++n)
        ldsB[n * LDSK + t] = wrow[n];
    __syncthreads();

    const int wave = t >> 5;
    const int lane = t & 31;
    const int half = lane >> 4;              // 0: K={k,k+1}, 1: K={k+2,k+3}
    const int lrow = lane & 15;              // matrix row (A) / col (B,C)

    const size_t tile_m = (size_t)blockIdx.x * 128 + (size_t)wave * 16;

    const float* Arow = A + (tile_m + lrow) * GEMM_K + 2 * half;   // A frag src
    const float* Brow = &ldsB[lrow * LDSK + 2 * half];             // B frag src

    v8f acc[NTILES] = {};
#pragma unroll 2
    for (int k0 = 0; k0 < GEMM_K; k0 += 4) {
        const v2f a = *(const v2f*)(Arow + k0);          // global_load_b64
#pragma unroll
        for (int j = 0; j < NTILES; ++j) {
            const v2f b = *(const v2f*)(Brow + j * 16 * LDSK + k0);  // ds_load_b64
            // D = A*B + C : v_wmma_f32_16x16x4_f32 (4 independent chains)
            acc[j] = __builtin_amdgcn_wmma_f32_16x16x4_f32(
                         false, a, false, b, (short)0, acc[j], false, false);
        }
    }

    // C/D layout: VGPR i -> row (i + 8*half), col lrow within each 16x16 tile.
#pragma unroll
    for (int j = 0; j < NTILES; ++j) {
        const int col = nb + j * 16 + lrow;
        const float bv = bias[col];
        float* Crow = C + (tile_m + 8 * half) * Ncols + col;
#pragma unroll
        for (int i = 0; i < 8; ++i)
            Crow[(size_t)i * Ncols] = acc[j][i] + bv;
    }
}

// ---------------------------------------------------------------------------
// Deformable sampling core (+ per-head softmax + location computation).
// grid.x = NQ_TOTAL (one block per (n,q)); 8 waves = 8 heads; lane = channel d.
// value:  (N, 21760, 8, 32) fp32   off_raw: (NQ, 256)   attn_raw: (NQ, 128)
// ref:    (N, Lq, 4, 2)            out acc: (NQ, 256)
// ---------------------------------------------------------------------------
__launch_bounds__(256)
__global__ void msda_sample(const float* __restrict__ value,
                            const float* __restrict__ ref,
                            const float* __restrict__ offr,
                            const float* __restrict__ attnr,
                            float* __restrict__ acc_out)
{
    const int nq   = blockIdx.x;             // n*Lq + q
    const int m    = threadIdx.x >> 5;       // head
    const int lane = threadIdx.x & 31;       // channel d
    const int n    = nq / LQ;

    // ---- softmax over the 16 (level,point) logits of this head --------------
    const float* ar = attnr + (size_t)nq * 128 + m * 16;
    float w[16];
    float mx = -3.0e38f;
#pragma unroll
    for (int i = 0; i < 16; ++i) { w[i] = ar[i]; mx = fmaxf(mx, w[i]); }
    float s = 0.0f;
#pragma unroll
    for (int i = 0; i < 16; ++i) { w[i] = __expf(w[i] - mx); s += w[i]; }
    const float inv_s = 1.0f / s;

    // ---- per-level constants ------------------------------------------------
    const int   Hs[4]  = {128, 64, 32, 16};
    const int   Ws[4]  = {128, 64, 32, 16};
    const int   st[4]  = {0, 16384, 20480, 21504};
    const float rH[4]  = {1.0f / 128.0f, 1.0f / 64.0f, 1.0f / 32.0f, 1.0f / 16.0f};

    const float* orow  = offr + (size_t)nq * 256;
    const float* rrow  = ref  + (size_t)nq * 8;
    const size_t vbase = ((size_t)n * LQ) * 256 + m * 32 + lane;

    float acc = 0.0f;
#pragma unroll
    for (int l = 0; l < 4; ++l) {
        const int   H = Hs[l], W = Ws[l], base = st[l];
        const float rx = rrow[l * 2 + 0];
        const float ry = rrow[l * 2 + 1];
#pragma unroll
        for (int p = 0; p < 4; ++p) {
            const int oi = ((m * 4 + l) * 4 + p) * 2;
            const float lx = rx + orow[oi + 0] * rH[l];     // norm = [W,H]; W==H here
            const float ly = ry + orow[oi + 1] * rH[l];
            const float x  = lx * (float)W - 0.5f;
            const float y  = ly * (float)H - 0.5f;
            const float x0 = floorf(x), y0 = floorf(y);
            const float fx = x - x0,    fy = y - y0;
            const int ix0 = (int)x0, iy0 = (int)y0;
            const int ix1 = ix0 + 1, iy1 = iy0 + 1;

            // zero-padding: zero the weight, clamp the index (always in-bounds)
            const float wx0 = (ix0 >= 0 && ix0 < W) ? (1.0f - fx) : 0.0f;
            const float wx1 = (ix1 >= 0 && ix1 < W) ? fx          : 0.0f;
            const float wy0 = (iy0 >= 0 && iy0 < H) ? (1.0f - fy) : 0.0f;
            const float wy1 = (iy1 >= 0 && iy1 < H) ? fy          : 0.0f;
            const int cx0 = min(max(ix0, 0), W - 1);
            const int cx1 = min(max(ix1, 0), W - 1);
            const int cy0 = min(max(iy0, 0), H - 1);
            const int cy1 = min(max(iy1, 0), H - 1);

            // coalesced 128B gathers (D innermost in value layout)
            const float v00 = value[vbase + ((size_t)(base + cy0 * W + cx0)) * 256];
            const float v01 = value[vbase + ((size_t)(base + cy0 * W + cx1)) * 256];
            const float v10 = value[vbase + ((size_t)(base + cy1 * W + cx0)) * 256];
            const float v11 = value[vbase + ((size_t)(base + cy1 * W + cx1)) * 256];

            const float samp = wy0 * (wx0 * v00 + wx1 * v01)
                             + wy1 * (wx0 * v10 + wx1 * v11);
            acc = fmaf(w[l * 4 + p], samp, acc);
        }
    }
    acc_out[(size_t)nq * 256 + m * 32 + lane] = acc * inv_s;
}

// ---------------------------------------------------------------------------
extern "C" void kernel_launch(void* const* d_in, const int* in_sizes, int n_in,
                              void* d_out, int out_size, void* d_ws, size_t ws_size,
                              hipStream_t stream)
{
    const float* query  = (const float*)d_in[0];
    const float* refpts = (const float*)d_in[1];
    const float* inflat = (const float*)d_in[2];
    const float* W_off  = (const float*)d_in[3];
    const float* b_off  = (const float*)d_in[4];
    const float* W_attn = (const float*)d_in[5];
    const float* b_attn = (const float*)d_in[6];
    const float* W_val  = (const float*)d_in[7];
    const float* b_val  = (const float*)d_in[8];
    const float* W_out  = (const float*)d_in[9];
    const float* b_out  = (const float*)d_in[10];
    float* out = (float*)d_out;

    // workspace layout (floats)
    float* ws      = (float*)d_ws;
    float* ws_val  = ws;                                  // 43520*256
    float* ws_off  = ws_val + (size_t)NQ_TOTAL * 256;     // 43520*256
    float* ws_attn = ws_off + (size_t)NQ_TOTAL * 256;     // 43520*128
    float* ws_acc  = ws_attn + (size_t)NQ_TOTAL * 128;    // 43520*256

    const dim3 blk(256);
    const int mblocks = NQ_TOTAL / 128;                   // 340

    // 1. value projection
    gemm_wmma_f32<<<dim3(mblocks, 4), blk, 0, stream>>>(inflat, W_val, b_val, ws_val, 256);
    // 2. sampling offsets
    gemm_wmma_f32<<<dim3(mblocks, 4), blk, 0, stream>>>(query, W_off, b_off, ws_off, 256);
    // 3. attention logits
    gemm_wmma_f32<<<dim3(mblocks, 2), blk, 0, stream>>>(query, W_attn, b_attn, ws_attn, 128);
    // 4. softmax + bilinear sampling + weighted accumulation
    msda_sample<<<dim3(NQ_TOTAL), blk, 0, stream>>>(ws_val, refpts, ws_off, ws_attn, ws_acc);
    // 5. output projection
    gemm_wmma_f32<<<dim3(mblocks, 4), blk, 0, stream>>>(ws_acc, W_out, b_out, out, 256);
}